// PointNetFeaturePropagation_712964571703
// MI455X (gfx1250) — compile-verified
//
#include <hip/hip_runtime.h>
#include <hip/hip_bf16.h>

// ---------------- problem constants ----------------
constexpr int BB  = 16;      // batch
constexpr int NN  = 4096;    // query points
constexpr int SS  = 1024;    // source points
constexpr int D1C = 128;     // points1 channels
constexpr int D2C = 256;     // points2 channels
constexpr int C0  = 384;     // concat channels (D1+D2)
constexpr int C1  = 256;     // MLP layer 1 out
constexpr int C2  = 128;     // MLP layer 2 out
constexpr int MM  = BB * NN; // 65536 rows
constexpr float BN_EPS = 1e-5f;

// reduction geometry
constexpr int NBLK = 256;        // partial-reduction blocks
constexpr int RPB  = MM / NBLK;  // rows per partial block = 256

// workspace layout (float offsets)
constexpr size_t OFF_X0   = 0;                                 // M x 384
constexpr size_t OFF_Y1   = OFF_X0 + (size_t)MM * C0;          // M x 256
constexpr size_t OFF_IDXW = OFF_Y1 + (size_t)MM * C1;          // M x 6
constexpr size_t OFF_P1   = OFF_IDXW + (size_t)MM * 6;         // NBLK x C1 x 2
constexpr size_t OFF_P2   = OFF_P1 + (size_t)NBLK * C1 * 2;    // NBLK x C2 x 2
constexpr size_t OFF_SB1  = OFF_P2 + (size_t)NBLK * C2 * 2;    // C1 x 2
constexpr size_t OFF_SB2  = OFF_SB1 + (size_t)C1 * 2;          // C2 x 2
constexpr size_t OFF_Y2   = 0;                                 // aliases X0 (dead)

typedef float v2f __attribute__((ext_vector_type(2)));
typedef float v8f __attribute__((ext_vector_type(8)));
typedef int   v4i __attribute__((ext_vector_type(4)));

// CDNA5 async global->LDS copy path (ASYNCcnt), with graceful fallback.
#if defined(__AMDGCN__) && __has_builtin(__builtin_amdgcn_global_load_async_to_lds_b128) && __has_builtin(__builtin_amdgcn_s_wait_asynccnt)
#define USE_ASYNC_LDS 1
typedef __attribute__((address_space(1))) v4i gv4i_t; // global
typedef __attribute__((address_space(3))) v4i lv4i_t; // LDS
#else
#define USE_ASYNC_LDS 0
#endif

// =====================================================================
// Kernel 1: 3-nearest-neighbour search + inverse-distance weights.
// One block per (b, 256-point chunk); xyz2[b] (12 KB) staged in LDS.
// =====================================================================
__global__ __launch_bounds__(256) void knn3_kernel(
    const float* __restrict__ xyz1, const float* __restrict__ xyz2,
    float* __restrict__ idxw)
{
    __shared__ float s2[3 * SS]; // 12 KB
    const int b = blockIdx.y;
    const int n = blockIdx.x * 256 + threadIdx.x;

    for (int i = threadIdx.x; i < 3 * SS; i += 256)
        s2[i] = xyz2[(size_t)b * 3 * SS + i];
    __syncthreads();

    const float px = xyz1[(size_t)b * 3 * NN + 0 * NN + n];
    const float py = xyz1[(size_t)b * 3 * NN + 1 * NN + n];
    const float pz = xyz1[(size_t)b * 3 * NN + 2 * NN + n];

    const float* sx = s2;
    const float* sy = s2 + SS;
    const float* sz = s2 + 2 * SS;

    float d0 = 3.4e38f, d1 = 3.4e38f, d2v = 3.4e38f;
    int   i0 = 0, i1 = 0, i2 = 0;

    #pragma unroll 4
    for (int s = 0; s < SS; ++s) {
        const float dx = px - sx[s];
        const float dy = py - sy[s];
        const float dz = pz - sz[s];
        const float d  = fmaf(dx, dx, fmaf(dy, dy, dz * dz));
        if (d < d2v) {
            if (d < d1) {
                if (d < d0) { d2v = d1; i2 = i1; d1 = d0; i1 = i0; d0 = d; i0 = s; }
                else        { d2v = d1; i2 = i1; d1 = d;  i1 = s; }
            } else          { d2v = d;  i2 = s; }
        }
    }

    const float r0 = 1.f / (d0 + 1e-8f);
    const float r1 = 1.f / (d1 + 1e-8f);
    const float r2 = 1.f / (d2v + 1e-8f);
    const float rs = 1.f / (r0 + r1 + r2);

    float* o = idxw + ((size_t)b * NN + n) * 6;
    o[0] = __int_as_float(i0);
    o[1] = __int_as_float(i1);
    o[2] = __int_as_float(i2);
    o[3] = r0 * rs;
    o[4] = r1 * rs;
    o[5] = r2 * rs;
}

// =====================================================================
// Kernel 2: build X0[m][0..383] = concat(points1[b,:,n], interp[b,:,n]).
// One block (384 threads) per point; row write fully coalesced; the
// channel-major gathers of points1/points2 hit L2 (1-2 MB per batch).
// =====================================================================
__global__ __launch_bounds__(384) void gather_concat_kernel(
    const float* __restrict__ p1, const float* __restrict__ p2,
    const float* __restrict__ idxw, float* __restrict__ X0)
{
    const int n = blockIdx.x;
    const int b = blockIdx.y;
    const size_t m = (size_t)b * NN + n;
    const int j = threadIdx.x;

    const float* iw = idxw + m * 6;
    float val;
    if (j < D1C) {
        val = p1[((size_t)b * D1C + j) * NN + n];
    } else {
        const int c  = j - D1C;
        const int i0 = __float_as_int(iw[0]);
        const int i1 = __float_as_int(iw[1]);
        const int i2 = __float_as_int(iw[2]);
        const float* base = p2 + ((size_t)b * D2C + c) * SS;
        val = fmaf(iw[3], base[i0], fmaf(iw[4], base[i1], iw[5] * base[i2]));
    }
    X0[m * C0 + j] = val;
}

// =====================================================================
// Kernel 3: fp32 WMMA GEMM  Y[M,NC] = X[M,K] * W[NC,K]^T + bias[NC]
// 256 threads (8 wave32), macro-tile 128x64, BK=16, double-buffered LDS.
// Tile staging uses GLOBAL_LOAD_ASYNC_TO_LDS_B128 (ASYNCcnt) when the
// toolchain exposes it; LDS pitch = 20 floats so every async B128 lands
// 16B-aligned and fragment ds_load_b64 reads stay bank-spread.
// Each wave: 32x32 tile = 2x2 v_wmma_f32_16x16x4_f32 accumulators.
// =====================================================================
template <int KDIM, int NCOLS>
__global__ __launch_bounds__(256) void gemm_wmma_kernel(
    const float* __restrict__ X, const float* __restrict__ W,
    const float* __restrict__ bias, float* __restrict__ Y)
{
    constexpr int BM = 128, BN = 64, BK = 16, LDK = BK + 4; // pitch 20 floats
    __shared__ float As[2][BM * LDK];
    __shared__ float Bs[2][BN * LDK];

    const int m0   = blockIdx.x * BM;
    const int n0   = blockIdx.y * BN;
    const int tid  = threadIdx.x;
    const int wave = tid >> 5;
    const int lane = tid & 31;
    const int wm   = (wave >> 1) * 32;  // wave M offset: 0,32,64,96
    const int wn   = (wave & 1) * 32;   // wave N offset: 0,32
    const int lr   = lane & 15;         // row within 16
    const int kh   = (lane >> 4) * 2;   // K sub-offset: 0 or 2

    v8f acc[2][2];
    #pragma unroll
    for (int i = 0; i < 2; ++i)
        #pragma unroll
        for (int j = 0; j < 2; ++j)
            #pragma unroll
            for (int r = 0; r < 8; ++r) acc[i][j][r] = 0.f;

    auto loadTile = [&](int kt, int buf) {
#if USE_ASYNC_LDS
        // A tile: 128x16 floats = 512 B128 transfers; 2 per thread.
        #pragma unroll
        for (int r2 = 0; r2 < 2; ++r2) {
            const int idx = tid + r2 * 256;
            const int r   = idx >> 2;
            const int c4  = (idx & 3) * 4;
            const float* g = X + (size_t)(m0 + r) * KDIM + kt * BK + c4;
            __builtin_amdgcn_global_load_async_to_lds_b128(
                (gv4i_t*)(const_cast<float*>(g)),
                (lv4i_t*)&As[buf][r * LDK + c4],
                0, 0);
        }
        // B tile: 64x16 floats = 256 B128 transfers; 1 per thread.
        {
            const int r  = tid >> 2;
            const int c4 = (tid & 3) * 4;
            const float* g = W + (size_t)(n0 + r) * KDIM + kt * BK + c4;
            __builtin_amdgcn_global_load_async_to_lds_b128(
                (gv4i_t*)(const_cast<float*>(g)),
                (lv4i_t*)&Bs[buf][r * LDK + c4],
                0, 0);
        }
#else
        // Fallback: global -> VGPR -> ds_store staging.
        #pragma unroll
        for (int r2 = 0; r2 < 2; ++r2) {
            const int idx = tid + r2 * 256;
            const int r   = idx >> 2;
            const int c4  = (idx & 3) * 4;
            const float4 v = *(const float4*)(X + (size_t)(m0 + r) * KDIM + kt * BK + c4);
            float* dst = &As[buf][r * LDK + c4];
            dst[0] = v.x; dst[1] = v.y; dst[2] = v.z; dst[3] = v.w;
        }
        {
            const int r  = tid >> 2;
            const int c4 = (tid & 3) * 4;
            const float4 v = *(const float4*)(W + (size_t)(n0 + r) * KDIM + kt * BK + c4);
            float* dst = &Bs[buf][r * LDK + c4];
            dst[0] = v.x; dst[1] = v.y; dst[2] = v.z; dst[3] = v.w;
        }
#endif
    };

    loadTile(0, 0);
#if USE_ASYNC_LDS
    __builtin_amdgcn_s_wait_asynccnt(0);
#endif
    __syncthreads();

    constexpr int KT = KDIM / BK;
    for (int kt = 0; kt < KT; ++kt) {
        if (kt + 1 < KT) loadTile(kt + 1, (kt + 1) & 1);
        const int buf = kt & 1;
        #pragma unroll
        for (int kk = 0; kk < 4; ++kk) {
            v2f af[2], bf[2];
            #pragma unroll
            for (int mt = 0; mt < 2; ++mt)
                af[mt] = *(const v2f*)&As[buf][(wm + mt * 16 + lr) * LDK + kk * 4 + kh];
            #pragma unroll
            for (int nt = 0; nt < 2; ++nt)
                bf[nt] = *(const v2f*)&Bs[buf][(wn + nt * 16 + lr) * LDK + kk * 4 + kh];
            #pragma unroll
            for (int mt = 0; mt < 2; ++mt)
                #pragma unroll
                for (int nt = 0; nt < 2; ++nt)
                    acc[mt][nt] = __builtin_amdgcn_wmma_f32_16x16x4_f32(
                        false, af[mt], false, bf[nt],
                        (short)0, acc[mt][nt], false, false);
        }
#if USE_ASYNC_LDS
        __builtin_amdgcn_s_wait_asynccnt(0);
#endif
        __syncthreads();
    }

    // Epilogue: D layout — VGPR r: lanes 0-15 row r, lanes 16-31 row r+8.
    const int rowBase = m0 + wm + ((lane >> 4) * 8);
    #pragma unroll
    for (int nt = 0; nt < 2; ++nt) {
        const int col = n0 + wn + nt * 16 + lr;
        const float bv = bias[col];
        #pragma unroll
        for (int mt = 0; mt < 2; ++mt) {
            #pragma unroll
            for (int r = 0; r < 8; ++r) {
                const int row = rowBase + mt * 16 + r;
                Y[(size_t)row * NCOLS + col] = acc[mt][nt][r] + bv;
            }
        }
    }
}

// =====================================================================
// Kernel 4: per-channel partial sums (sum, sumsq) over RPB rows/block.
// Deterministic (fixed accumulation order per partial slot).
// =====================================================================
template <int C>
__global__ __launch_bounds__(256) void bn_stats_partial_kernel(
    const float* __restrict__ Y, float* __restrict__ partials)
{
    constexpr int TPC = 256 / C; // threads sharing a channel (1 or 2)
    __shared__ float red[512];
    const int t  = threadIdx.x;
    const int c  = t & (C - 1);
    const int rh = t / C;
    const int row0 = blockIdx.x * RPB;

    float s = 0.f, ss = 0.f;
    for (int r = rh; r < RPB; r += TPC) {
        const float v = Y[(size_t)(row0 + r) * C + c];
        s += v;
        ss = fmaf(v, v, ss);
    }
    if (TPC > 1) {
        red[t] = s; red[256 + t] = ss;
        __syncthreads();
        if (t < C) { s += red[t + C]; ss += red[256 + t + C]; }
    }
    if (t < C) {
        partials[((size_t)blockIdx.x * C + c) * 2 + 0] = s;
        partials[((size_t)blockIdx.x * C + c) * 2 + 1] = ss;
    }
}

// =====================================================================
// Kernel 5: finalize BN: per-channel tree-reduce NBLK partials, emit
// fused scale = g*rsqrt(var+eps), bias' = beta - mean*scale.
// =====================================================================
template <int C>
__global__ __launch_bounds__(256) void bn_stats_final_kernel(
    const float* __restrict__ partials, const float* __restrict__ g,
    const float* __restrict__ beta, float* __restrict__ sb)
{
    __shared__ float rs[256], rss[256];
    const int c = blockIdx.x;
    const int t = threadIdx.x; // == NBLK
    rs[t]  = partials[((size_t)t * C + c) * 2 + 0];
    rss[t] = partials[((size_t)t * C + c) * 2 + 1];
    __syncthreads();
    for (int o = 128; o > 0; o >>= 1) {
        if (t < o) { rs[t] += rs[t + o]; rss[t] += rss[t + o]; }
        __syncthreads();
    }
    if (t == 0) {
        const float inv   = 1.f / (float)MM;
        const float mean  = rs[0] * inv;
        const float var   = rss[0] * inv - mean * mean;
        const float scale = g[c] * rsqrtf(var + BN_EPS);
        sb[c * 2 + 0] = scale;
        sb[c * 2 + 1] = beta[c] - mean * scale;
    }
}

// =====================================================================
// Kernel 6: in-place fused BN + ReLU (float4 vectorized).
// =====================================================================
template <int C>
__global__ __launch_bounds__(256) void bn_relu_apply_kernel(
    float* __restrict__ Y, const float* __restrict__ sb)
{
    const size_t i4 = (size_t)blockIdx.x * 256 + threadIdx.x;
    const int c0 = (int)((i4 * 4) & (C - 1));
    float4 v = ((float4*)Y)[i4];
    v.x = fmaxf(0.f, fmaf(v.x, sb[2 * (c0 + 0)], sb[2 * (c0 + 0) + 1]));
    v.y = fmaxf(0.f, fmaf(v.y, sb[2 * (c0 + 1)], sb[2 * (c0 + 1) + 1]));
    v.z = fmaxf(0.f, fmaf(v.z, sb[2 * (c0 + 2)], sb[2 * (c0 + 2) + 1]));
    v.w = fmaxf(0.f, fmaf(v.w, sb[2 * (c0 + 3)], sb[2 * (c0 + 3) + 1]));
    ((float4*)Y)[i4] = v;
}

// =====================================================================
// Kernel 7: fused BN + ReLU + transpose to output layout (B, C2, N).
// 32x32 LDS tile; both global sides coalesced.
// =====================================================================
__global__ __launch_bounds__(256) void bn_relu_transpose_kernel(
    const float* __restrict__ Y, const float* __restrict__ sb,
    float* __restrict__ out)
{
    __shared__ float tile[32][33];
    const int b  = blockIdx.z;
    const int n0 = blockIdx.x * 32;
    const int c0 = blockIdx.y * 32;
    const int tx = threadIdx.x; // 32
    const int ty = threadIdx.y; // 8

    const float sc = sb[2 * (c0 + tx) + 0];
    const float bi = sb[2 * (c0 + tx) + 1];

    #pragma unroll
    for (int i = 0; i < 4; ++i) {
        const int nn = ty + i * 8;
        const float v = Y[((size_t)(b * NN + n0 + nn)) * C2 + c0 + tx];
        tile[nn][tx] = fmaxf(0.f, fmaf(v, sc, bi));
    }
    __syncthreads();
    #pragma unroll
    for (int i = 0; i < 4; ++i) {
        const int cc = ty + i * 8;
        out[((size_t)b * C2 + c0 + cc) * NN + n0 + tx] = tile[tx][cc];
    }
}

// =====================================================================
extern "C" void kernel_launch(void* const* d_in, const int* in_sizes, int n_in,
                              void* d_out, int out_size, void* d_ws, size_t ws_size,
                              hipStream_t stream)
{
    const float* xyz1  = (const float*)d_in[0];
    const float* xyz2  = (const float*)d_in[1];
    const float* pts1  = (const float*)d_in[2];
    const float* pts2  = (const float*)d_in[3];
    const float* w1    = (const float*)d_in[4];
    const float* b1    = (const float*)d_in[5];
    const float* g1    = (const float*)d_in[6];
    const float* beta1 = (const float*)d_in[7];
    const float* w2    = (const float*)d_in[8];
    const float* b2    = (const float*)d_in[9];
    const float* g2    = (const float*)d_in[10];
    const float* beta2 = (const float*)d_in[11];
    float* out = (float*)d_out;

    float* ws   = (float*)d_ws;
    float* X0   = ws + OFF_X0;
    float* Y1   = ws + OFF_Y1;
    float* Y2   = ws + OFF_Y2;   // aliases X0 (X0 dead after gemm1)
    float* IDXW = ws + OFF_IDXW;
    float* P1S  = ws + OFF_P1;
    float* P2S  = ws + OFF_P2;
    float* SB1  = ws + OFF_SB1;
    float* SB2  = ws + OFF_SB2;

    // 1) 3-NN + weights
    knn3_kernel<<<dim3(NN / 256, BB), 256, 0, stream>>>(xyz1, xyz2, IDXW);

    // 2) gather + concat -> X0 [M,384]
    gather_concat_kernel<<<dim3(NN, BB), 384, 0, stream>>>(pts1, pts2, IDXW, X0);

    // 3) layer 1 GEMM: Y1 = X0 * w1^T + b1
    gemm_wmma_kernel<C0, C1><<<dim3(MM / 128, C1 / 64), 256, 0, stream>>>(X0, w1, b1, Y1);

    // 4) BN1 stats + apply (fused with ReLU), in place on Y1
    bn_stats_partial_kernel<C1><<<NBLK, 256, 0, stream>>>(Y1, P1S);
    bn_stats_final_kernel<C1><<<C1, 256, 0, stream>>>(P1S, g1, beta1, SB1);
    bn_relu_apply_kernel<C1><<<(size_t)MM * C1 / 4 / 256, 256, 0, stream>>>(Y1, SB1);

    // 5) layer 2 GEMM: Y2 = Y1 * w2^T + b2
    gemm_wmma_kernel<C1, C2><<<dim3(MM / 128, C2 / 64), 256, 0, stream>>>(Y1, w2, b2, Y2);

    // 6) BN2 stats, then fused BN+ReLU+transpose into (B, C2, N) output
    bn_stats_partial_kernel<C2><<<NBLK, 256, 0, stream>>>(Y2, P2S);
    bn_stats_final_kernel<C2><<<C2, 256, 0, stream>>>(P2S, g2, beta2, SB2);
    bn_relu_transpose_kernel<<<dim3(NN / 32, C2 / 32, BB), dim3(32, 8), 0, stream>>>(Y2, SB2, out);
}